// GINGCNLayer_86148454023364
// MI455X (gfx1250) — compile-verified
//
#include <hip/hip_runtime.h>

typedef long long i64;

#define TPB 256

// ===========================================================================
// Fast path: bucketed (counting-sort) SpMM with premultiplied edge weights.
// ~3.3M cheap atomics; heavy phase is atomic-free wave32 register accumulation
// over coalesced L2 gathers with cross-lane index broadcast.
// ===========================================================================

// --- 1) zero counters ------------------------------------------------------
__global__ void k_init_ws(float* __restrict__ degf, int* __restrict__ cnt,
                          unsigned int* __restrict__ total, int n) {
    int i = blockIdx.x * blockDim.x + threadIdx.x;
    if (i < n) {
        degf[i] = 0.0f;
        cnt[i]  = 0;
    }
    if (i == 0) total[0] = 0u;
}

// --- 2) edge histograms: in-degree by col (f32) + bucket sizes by row ------
__global__ void k_hist(const i64* __restrict__ row, const i64* __restrict__ col,
                       float* __restrict__ degf, int* __restrict__ cnt, int nE) {
    int e = blockIdx.x * blockDim.x + threadIdx.x;
    if (e < nE) {
        int c = (int)col[e];
        int r = (int)row[e];
        unsafeAtomicAdd(&degf[c], 1.0f);   // global_atomic_add_f32, no return
        atomicAdd(&cnt[r], 1);             // global_atomic_add_u32
    }
}

// --- 3) per node: dis = deg^{-1/2} + segment allocation (block-aggregated
//        cursor: LDS scan + ONE same-address atomic per block) --------------
__global__ void k_node_setup(float* __restrict__ degf, const int* __restrict__ cnt,
                             int* __restrict__ off, int* __restrict__ cursor,
                             unsigned int* __restrict__ total, int n) {
    __shared__ int s[TPB];
    __shared__ int base;
    int i = blockIdx.x * blockDim.x + threadIdx.x;
    int v = (i < n) ? cnt[i] : 0;
    s[threadIdx.x] = v;
    __syncthreads();
    for (int d = 1; d < TPB; d <<= 1) {        // Hillis-Steele inclusive scan
        int t = (threadIdx.x >= (unsigned)d) ? s[threadIdx.x - d] : 0;
        __syncthreads();
        s[threadIdx.x] += t;
        __syncthreads();
    }
    if (threadIdx.x == TPB - 1)
        base = (int)atomicAdd(total, (unsigned int)s[TPB - 1]);
    __syncthreads();
    int excl = s[threadIdx.x] - v;
    if (i < n) {
        int o = base + excl;
        off[i]    = o;
        cursor[i] = o;
        float d0  = degf[i];
        degf[i]   = (d0 > 0.0f) ? __frsqrt_rn(d0) : 0.0f;   // degf[] is now dis[]
    }
}

// --- 4) bucket edges as 8B (col, weight) pairs; weight premultiplied so the
//        hot loop has NO dependent dis[] gather ------------------------------
__global__ void k_bucket(const i64* __restrict__ row, const i64* __restrict__ col,
                         const float* __restrict__ dis,
                         int* __restrict__ cursor, uint2* __restrict__ pairs,
                         int nE) {
    int e = blockIdx.x * blockDim.x + threadIdx.x;
    if (e < nE) {
        int r = (int)row[e];
        int c = (int)col[e];
        float w = dis[r] * dis[c];
        int p = atomicAdd(&cursor[r], 1);
        pairs[p] = make_uint2((unsigned int)c, __float_as_uint(w));
    }
}

// --- 5) atomic-free aggregation: one wave32 per node.
//        * 32 lanes load up to 32 (col,w) pairs in ONE coalesced b64 load
//        * inner loop broadcasts pair j cross-lane (__shfl -> ds_bpermute /
//          v_readlane), gathers the 512B feature row coalesced (b128/lane),
//          prefetches 4 neighbors ahead (global_prefetch_b8), FMAs into a
//          float4 accumulator.  Self term (1+eps)*x fused.  One b128 store. --
__global__ void k_aggregate(const float* __restrict__ x,
                            const uint2* __restrict__ pairs,
                            const int* __restrict__ off,
                            const int* __restrict__ cnt,
                            const float* __restrict__ eps,
                            float* __restrict__ out, int nN) {
    const int lane = threadIdx.x & 31;
    const int node = blockIdx.x * (blockDim.x >> 5) + (threadIdx.x >> 5);
    if (node >= nN) return;

    const float s = 1.0f + eps[0];
    const float4 xv = *(const float4*)(x + (size_t)node * 128 + lane * 4);
    float4 acc = make_float4(s * xv.x, s * xv.y, s * xv.z, s * xv.w);

    const int o = off[node];
    const int n = cnt[node];

    for (int b = 0; b < n; b += 32) {
        const int m = (n - b < 32) ? (n - b) : 32;
        uint2 p = make_uint2(0u, 0u);
        if (lane < m) p = pairs[o + b + lane];      // one coalesced b64 / wave
        const int   myc = (int)p.x;
        const float myw = __uint_as_float(p.y);

        for (int j = 0; j < m; ++j) {
            const int   c = __shfl(myc, j, 32);     // cross-lane broadcast
            const float w = __shfl(myw, j, 32);
            if (j + 4 < m) {
                int c4 = __shfl(myc, j + 4, 32);
                __builtin_prefetch(x + (size_t)c4 * 128 + lane * 4, 0, 3);
            }
            const float4 v = *(const float4*)(x + (size_t)c * 128 + lane * 4);
            acc.x = fmaf(w, v.x, acc.x);
            acc.y = fmaf(w, v.y, acc.y);
            acc.z = fmaf(w, v.z, acc.z);
            acc.w = fmaf(w, v.w, acc.w);
        }
    }
    *(float4*)(out + (size_t)node * 128 + lane * 4) = acc;
}

// ===========================================================================
// Fallback path (small workspace): direct atomic scatter
// ===========================================================================
__global__ void k_zero_f32(float* __restrict__ p, int n) {
    int i = blockIdx.x * blockDim.x + threadIdx.x;
    if (i < n) p[i] = 0.0f;
}

__global__ void k_degree_only(const i64* __restrict__ col, float* __restrict__ deg,
                              int nE) {
    int e = blockIdx.x * blockDim.x + threadIdx.x;
    if (e < nE) unsafeAtomicAdd(&deg[(int)col[e]], 1.0f);
}

__global__ void k_inv_sqrt(float* __restrict__ deg, int n) {
    int i = blockIdx.x * blockDim.x + threadIdx.x;
    if (i < n) {
        float d = deg[i];
        deg[i] = (d > 0.0f) ? __frsqrt_rn(d) : 0.0f;
    }
}

__global__ void k_init_out(const float4* __restrict__ x4, const float* __restrict__ eps,
                           float4* __restrict__ out4, int n4) {
    int i = blockIdx.x * blockDim.x + threadIdx.x;
    if (i < n4) {
        float s = 1.0f + eps[0];
        float4 v = x4[i];
        out4[i] = make_float4(s * v.x, s * v.y, s * v.z, s * v.w);
    }
}

__global__ void k_scatter_atomic(const float* __restrict__ x,
                                 const i64* __restrict__ row,
                                 const i64* __restrict__ col,
                                 const float* __restrict__ dis,
                                 float* __restrict__ out, int nE) {
    const int lane = threadIdx.x & 31;
    const int wid  = blockIdx.x * (blockDim.x >> 5) + (threadIdx.x >> 5);
    const int nw   = gridDim.x * (blockDim.x >> 5);
    for (int e = wid; e < nE; e += nw) {
        int r = (int)row[e];
        int c = (int)col[e];
        float norm = dis[r] * dis[c];
        const float4 v = *(const float4*)(x + (size_t)c * 128 + lane * 4);
        float* o = out + (size_t)r * 128 + lane * 4;
        unsafeAtomicAdd(o + 0, norm * v.x);
        unsafeAtomicAdd(o + 1, norm * v.y);
        unsafeAtomicAdd(o + 2, norm * v.z);
        unsafeAtomicAdd(o + 3, norm * v.w);
    }
}

// ===========================================================================
// host side
// ===========================================================================
extern "C" void kernel_launch(void* const* d_in, const int* in_sizes, int n_in,
                              void* d_out, int out_size, void* d_ws, size_t ws_size,
                              hipStream_t stream) {
    const float* x   = (const float*)d_in[0];   // [N,128] f32
    const float* eps = (const float*)d_in[1];   // [1] f32
    const i64*   ei  = (const i64*)d_in[2];     // [2,E] int64 (reference dtype)
    float*       out = (float*)d_out;

    const int nE = in_sizes[2] / 2;
    const int nN = in_sizes[0] / 128;
    const i64* row = ei;
    const i64* col = ei + nE;

    const int nodeBlocks = (nN + TPB - 1) / TPB;
    const int edgeBlocks = (nE + TPB - 1) / TPB;

    const size_t need = (size_t)nN * 16 + 64 + (size_t)nE * 8;

    if (ws_size >= need) {
        float* degf          = (float*)d_ws;                 // nN f32 -> dis[]
        int*   cnt           = (int*)(degf + nN);            // nN i32
        int*   off           = cnt + nN;                     // nN i32
        int*   cursor        = off + nN;                     // nN i32
        unsigned int* total  = (unsigned int*)(cursor + nN); // 16 ints (pad)
        uint2* pairs         = (uint2*)(total + 16);         // nE x 8B

        k_init_ws<<<nodeBlocks, TPB, 0, stream>>>(degf, cnt, total, nN);
        k_hist<<<edgeBlocks, TPB, 0, stream>>>(row, col, degf, cnt, nE);
        k_node_setup<<<nodeBlocks, TPB, 0, stream>>>(degf, cnt, off, cursor, total, nN);
        k_bucket<<<edgeBlocks, TPB, 0, stream>>>(row, col, degf, cursor, pairs, nE);

        const int waveBlocks = (nN + (TPB >> 5) - 1) / (TPB >> 5);  // 8 nodes/block
        k_aggregate<<<waveBlocks, TPB, 0, stream>>>(x, pairs, off, cnt, eps, out, nN);
    } else {
        float* deg = (float*)d_ws;
        k_zero_f32<<<nodeBlocks, TPB, 0, stream>>>(deg, nN);
        k_degree_only<<<edgeBlocks, TPB, 0, stream>>>(col, deg, nE);
        k_inv_sqrt<<<nodeBlocks, TPB, 0, stream>>>(deg, nN);
        const int n4 = in_sizes[0] / 4;
        k_init_out<<<(n4 + TPB - 1) / TPB, TPB, 0, stream>>>(
            (const float4*)x, eps, (float4*)out, n4);
        k_scatter_atomic<<<2048, TPB, 0, stream>>>(x, row, col, deg, out, nE);
    }
}